// Loss_62929860821403
// MI455X (gfx1250) — compile-verified
//
#include <hip/hip_runtime.h>
#include <hip/hip_bf16.h>

typedef __attribute__((ext_vector_type(2))) float v2f;
typedef __attribute__((ext_vector_type(8))) float v8f;

#define B_ROWS 262144
#define D_DIM  256
#define C_CLS  10
#define CSTR   258   // padded LDS stride (floats) for center tile -> no bank conflicts

// workspace layout (floats)
#define WS_INTRA 0
#define WS_SUMS  64
#define WS_CNTS  (64 + C_CLS * D_DIM)
#define WS_TOTAL (WS_CNTS + 16)

__global__ __launch_bounds__(256) void k_init(float* __restrict__ w) {
    int i = blockIdx.x * 256 + threadIdx.x;
    if (i < WS_TOTAL) w[i] = 0.f;
}

// ---- Pass 1: intra loss via fp32 WMMA (F . C^T), expansion d2 = aa + bb - 2*dot ----
__global__ __launch_bounds__(256) void k_intra(const float* __restrict__ feat,
                                               const int*   __restrict__ labels,
                                               const float* __restrict__ center,
                                               float* __restrict__ w) {
    __shared__ float ldsC[16 * CSTR];     // padded 16x256 center tile (rows 10..15 = 0)
    __shared__ float ldsTile[8][16 * 16]; // per-wave D-matrix spill
    __shared__ float ldsAA[8][16];        // per-wave row norms
    __shared__ float ldsBB[16];           // center norms
    __shared__ float ldsIntra;

    const int tid   = threadIdx.x;
    const int wave  = tid >> 5;
    const int lane  = tid & 31;
    const int mrow  = lane & 15;   // A-matrix M / B-matrix N for this lane
    const int halfk = lane >> 4;   // which K-pair of the 4-wide K step

    if (tid == 0) ldsIntra = 0.f;
    for (int i = tid; i < 16 * D_DIM; i += 256) {
        int r = i >> 8, c = i & 255;
        ldsC[r * CSTR + c] = (r < C_CLS) ? center[r * D_DIM + c] : 0.f;
    }
    __syncthreads();
    if (tid < 16) {
        float s = 0.f;
        for (int d = 0; d < D_DIM; ++d) { float v = ldsC[tid * CSTR + d]; s += v * v; }
        ldsBB[tid] = s;   // zero for padded rows
    }
    __syncthreads();

    const int    rowBase = (blockIdx.x * 8 + wave) * 16;
    const float* fptr    = feat + (size_t)(rowBase + mrow) * D_DIM;
    const float* bptr    = &ldsC[mrow * CSTR];

    v8f   acc     = {};
    float aa_part = 0.f;
    #pragma unroll 4
    for (int kk = 0; kk < 64; ++kk) {
        const int k = kk * 4 + halfk * 2;
        v2f a = *(const v2f*)(fptr + k);   // A: lane holds M=mrow, K={k,k+1}
        v2f b = *(const v2f*)(bptr + k);   // B: lane holds N=mrow, K={k,k+1}
        aa_part += a.x * a.x + a.y * a.y;
        // D = A(16x4) * B(4x16) + C  -- full fp32 matrix op
        acc = __builtin_amdgcn_wmma_f32_16x16x4_f32(
            /*neg_a=*/false, a, /*neg_b=*/false, b,
            /*c_mod=*/(short)0, acc, /*reuse_a=*/false, /*reuse_b=*/false);
    }

    // aa[m] = lane m part (K%4 in {0,1}) + lane m+16 part (K%4 in {2,3})
    float aa = aa_part + __shfl_xor(aa_part, 16, 32);
    if (lane < 16) ldsAA[wave][mrow] = aa;
    #pragma unroll
    for (int j = 0; j < 8; ++j)  // D VGPR j: M = j + 8*halfk, N = mrow
        ldsTile[wave][(j + 8 * halfk) * 16 + mrow] = acc[j];
    __syncthreads();

    if (lane < 16) {
        const int   row = rowBase + mrow;
        const int   l   = labels[row];
        float d2 = ldsAA[wave][mrow] + ldsBB[l] - 2.f * ldsTile[wave][mrow * 16 + l];
        d2 = fmaxf(d2, 0.f);
        float dist = sqrtf(d2);
        dist = fminf(fmaxf(dist, 1e-16f), 1e16f);
        atomicAdd(&ldsIntra, dist);
    }
    __syncthreads();
    if (tid == 0) atomicAdd(&w[WS_INTRA], ldsIntra);
}

// ---- Pass 2: per-class feature sums + counts, LDS-privatized ----
__global__ __launch_bounds__(256) void k_segsum(const float* __restrict__ feat,
                                                const int*   __restrict__ labels,
                                                float* __restrict__ w) {
    __shared__ float lsum[C_CLS * D_DIM];
    __shared__ int   lcnt[C_CLS];
    const int tid = threadIdx.x;
    for (int i = tid; i < C_CLS * D_DIM; i += 256) lsum[i] = 0.f;
    if (tid < C_CLS) lcnt[tid] = 0;
    __syncthreads();

    const int base = blockIdx.x * 256;
    for (int r = 0; r < 256; ++r) {
        const int row = base + r;
        const int l   = labels[row];                       // uniform across block
        lsum[l * D_DIM + tid] += feat[(size_t)row * D_DIM + tid];  // thread owns column tid
        if (tid == 0) lcnt[l] += 1;
    }
    __syncthreads();
    for (int i = tid; i < C_CLS * D_DIM; i += 256) atomicAdd(&w[WS_SUMS + i], lsum[i]);
    if (tid < C_CLS) atomicAdd(&w[WS_CNTS + tid], (float)lcnt[tid]);
}

// ---- Pass 3: finalize (single block) ----
__global__ __launch_bounds__(256) void k_final(const float* __restrict__ center,
                                               const float* __restrict__ w,
                                               float* __restrict__ out) {
    __shared__ float red[256];
    const int tid = threadIdx.x;

    const float inv8 = 1.f / fmaxf(w[WS_CNTS + C_CLS - 2], 1.f);
    const float inv9 = 1.f / fmaxf(w[WS_CNTS + C_CLS - 1], 1.f);
    const float a = (center[(C_CLS - 2) * D_DIM + tid] + w[WS_SUMS + (C_CLS - 2) * D_DIM + tid]) * inv8;
    const float b = (center[(C_CLS - 1) * D_DIM + tid] + w[WS_SUMS + (C_CLS - 1) * D_DIM + tid]) * inv9;
    const float diff = a - b;
    red[tid] = diff * diff;
    __syncthreads();
    for (int s = 128; s > 0; s >>= 1) {
        if (tid < s) red[tid] += red[tid + s];
        __syncthreads();
    }
    if (tid == 0) {
        const float last = sqrtf(fmaxf(red[0], 0.f));
        const float den  = fminf(fmaxf(2.f * last, 1e-16f), 1e16f);
        out[1] = 1.f / den;                       // inter
        out[0] = w[WS_INTRA] / (float)B_ROWS;     // intra mean
    }
    for (int i = tid; i < C_CLS * D_DIM; i += 256) out[2 + i] = center[i];  // pass-through
}

extern "C" void kernel_launch(void* const* d_in, const int* in_sizes, int n_in,
                              void* d_out, int out_size, void* d_ws, size_t ws_size,
                              hipStream_t stream) {
    const float* feat   = (const float*)d_in[0];
    const int*   labels = (const int*)d_in[1];
    const float* center = (const float*)d_in[2];
    float*       out    = (float*)d_out;
    float*       w      = (float*)d_ws;
    (void)in_sizes; (void)n_in; (void)out_size; (void)ws_size;

    k_init  <<<(WS_TOTAL + 255) / 256, 256, 0, stream>>>(w);
    k_intra <<<B_ROWS / 128,          256, 0, stream>>>(feat, labels, center, w);
    k_segsum<<<B_ROWS / 256,          256, 0, stream>>>(feat, labels, w);
    k_final <<<1,                     256, 0, stream>>>(center, w, out);
}